// Attn_23252952940844
// MI455X (gfx1250) — compile-verified
//
#include <hip/hip_runtime.h>
#include <math.h>

typedef __attribute__((ext_vector_type(2))) float v2f;
typedef __attribute__((ext_vector_type(4))) float v4f;
typedef __attribute__((ext_vector_type(8))) float v8f;

#define S_LEN 2048
#define B_SZ  32
#define H_SZ  1024

// ---------------------------------------------------------------------------
// Kernel 1: u = hidden @ W   (M=32, N=1024, K=1024, all f32)
// u[m,n] = sum_k hidden[m,k] * W[k,n]     (W row-major [k][n])
// One wave per 16x16 output tile, V_WMMA_F32_16X16X4_F32 over 256 k-steps.
//
// ISA 7.12.2 layouts (wave32):
//   A 16x4 (MxK): lanes 0-15 -> M=lane, VGPR0=K0, VGPR1=K1
//                 lanes 16-31 -> M=lane-16, VGPR0=K2, VGPR1=K3
//   B 4x16 (KxN): VGPR v, lanes 0-15 -> K=v,   N=lane
//                          lanes 16-31 -> K=v+2, N=lane-16
//   C/D 16x16:    VGPR v, lanes 0-15 -> M=v,   N=lane
//                          lanes 16-31 -> M=v+8, N=lane-16
// ---------------------------------------------------------------------------
__global__ __launch_bounds__(32) void wmma_u_kernel(const float* __restrict__ hidden,
                                                    const float* __restrict__ W,
                                                    float* __restrict__ u) {
    const int n0   = blockIdx.x * 16;   // 64 tiles in N
    const int m0   = blockIdx.y * 16;   // 2 tiles in M
    const int lane = threadIdx.x;       // 0..31
    const int half = lane >> 4;         // 0 = lanes 0-15, 1 = lanes 16-31
    const int l16  = lane & 15;

    const float* arow = hidden + (size_t)(m0 + l16) * H_SZ;  // A row for this lane
    v8f acc = {};

    for (int k = 0; k < H_SZ; k += 4) {
        const int ka = k + 2 * half;     // K pair selected by lane half
        v2f a, b;
        a.x = arow[ka + 0];
        a.y = arow[ka + 1];
        b.x = W[(size_t)(ka + 0) * H_SZ + n0 + l16];
        b.y = W[(size_t)(ka + 1) * H_SZ + n0 + l16];
        // 8 args: (neg_a, A, neg_b, B, c_mod, C, reuse_a, reuse_b)
        acc = __builtin_amdgcn_wmma_f32_16x16x4_f32(false, a, false, b,
                                                    (short)0, acc, false, false);
    }

#pragma unroll
    for (int v = 0; v < 8; ++v) {
        u[(size_t)(m0 + v + 8 * half) * H_SZ + n0 + l16] = acc[v];
    }
}

// ---------------------------------------------------------------------------
// Kernel 2: attn[b,s] = u[b,:] . enc[s,b,:]
// Pure HBM-streaming kernel: 256 MB of enc read exactly once (NT loads).
// Block = 256 threads (8 waves). blockIdx = (s-chunk of 64, b).
// Each wave: keeps u[b,:] in 32 VGPRs, processes 8 s-rows.
// ---------------------------------------------------------------------------
__global__ __launch_bounds__(256) void score_kernel(const float* __restrict__ enc,
                                                    const float* __restrict__ u,
                                                    float* __restrict__ attn) {
    const int b     = blockIdx.y;              // 0..31
    const int sBase = blockIdx.x * 64;
    const int lane  = threadIdx.x & 31;
    const int wave  = threadIdx.x >> 5;        // 0..7

    // u[b,:] -> 8 float4 per lane (coalesced 16B/lane)
    v4f ureg[8];
    const v4f* urow = (const v4f*)(u + (size_t)b * H_SZ);
#pragma unroll
    for (int j = 0; j < 8; ++j) ureg[j] = urow[j * 32 + lane];

#pragma unroll
    for (int r = 0; r < 8; ++r) {
        const int s = sBase + wave * 8 + r;
        const v4f* erow = (const v4f*)(enc + ((size_t)s * B_SZ + b) * H_SZ);
        float acc = 0.0f;
#pragma unroll
        for (int j = 0; j < 8; ++j) {
            v4f e = __builtin_nontemporal_load(&erow[j * 32 + lane]);
            acc += e.x * ureg[j].x + e.y * ureg[j].y
                 + e.z * ureg[j].z + e.w * ureg[j].w;
        }
        // wave32 all-reduce
#pragma unroll
        for (int off = 16; off > 0; off >>= 1)
            acc += __shfl_xor(acc, off, 32);
        if (lane == 0) attn[(size_t)b * S_LEN + s] = acc;
    }
}

// ---------------------------------------------------------------------------
// Kernel 3: row softmax over S=2048 per batch element.
// (The hidden·bias constant cancels under softmax, so it was never computed.)
// ---------------------------------------------------------------------------
__global__ __launch_bounds__(256) void softmax_kernel(const float* __restrict__ attn,
                                                      float* __restrict__ out) {
    const int b   = blockIdx.x;
    const int tid = threadIdx.x;
    __shared__ float smem[8];

    float vals[8];
    float m = -INFINITY;
#pragma unroll
    for (int j = 0; j < 8; ++j) {
        vals[j] = attn[(size_t)b * S_LEN + j * 256 + tid];
        m = fmaxf(m, vals[j]);
    }
#pragma unroll
    for (int off = 16; off > 0; off >>= 1) m = fmaxf(m, __shfl_xor(m, off, 32));
    if ((tid & 31) == 0) smem[tid >> 5] = m;
    __syncthreads();
    float bm = smem[0];
#pragma unroll
    for (int w = 1; w < 8; ++w) bm = fmaxf(bm, smem[w]);
    __syncthreads();

    float sum = 0.0f;
#pragma unroll
    for (int j = 0; j < 8; ++j) { vals[j] = __expf(vals[j] - bm); sum += vals[j]; }
#pragma unroll
    for (int off = 16; off > 0; off >>= 1) sum += __shfl_xor(sum, off, 32);
    if ((tid & 31) == 0) smem[tid >> 5] = sum;
    __syncthreads();
    float bs = 0.0f;
#pragma unroll
    for (int w = 0; w < 8; ++w) bs += smem[w];
    const float inv = 1.0f / bs;
#pragma unroll
    for (int j = 0; j < 8; ++j)
        out[(size_t)b * S_LEN + j * 256 + tid] = vals[j] * inv;
}

// ---------------------------------------------------------------------------
extern "C" void kernel_launch(void* const* d_in, const int* in_sizes, int n_in,
                              void* d_out, int out_size, void* d_ws, size_t ws_size,
                              hipStream_t stream) {
    const float* hidden = (const float*)d_in[0];   // [B,H]
    const float* enc    = (const float*)d_in[1];   // [S,B,H]
    const float* W      = (const float*)d_in[2];   // [H,H], W[k,h] at k*H+h
    // d_in[3] = bias: unused — softmax is invariant to the per-b shift hidden·bias.

    float* u    = (float*)d_ws;                    // 32*1024 f32 = 128 KB
    float* attn = u + (size_t)B_SZ * H_SZ;         // 32*2048 f32 = 256 KB
    float* out  = (float*)d_out;                   // [1,B,S] = B*S f32

    dim3 g1(H_SZ / 16, B_SZ / 16);                 // 64 x 2 tiles
    wmma_u_kernel<<<g1, 32, 0, stream>>>(hidden, W, u);

    dim3 g2(S_LEN / 64, B_SZ);                     // 32 x 32 = 1024 blocks
    score_kernel<<<g2, 256, 0, stream>>>(enc, u, attn);

    softmax_kernel<<<B_SZ, 256, 0, stream>>>(attn, out);
}